// AdaptiveMHFConv_46231027974333
// MI455X (gfx1250) — compile-verified
//
#include <hip/hip_runtime.h>
#include <hip/hip_bf16.h>
#include <math.h>

// ---------------------------------------------------------------------------
// CDNA5 (gfx1250) WMMA f32 16x16x4 helpers.
// A-matrix 16x4: lanes 0-15 -> M=lane, VGPR0=K0,VGPR1=K1; lanes 16-31 -> K2,K3.
// B-matrix 4x16: symmetric (N=lane&15, K halves split across lane groups).
// C/D 16x16: VGPR v -> M=v (lanes 0-15) / M=v+8 (lanes 16-31), N=lane&15.
//
// All K-major operands use a K-pair-interleaved layout
//     addr(m, k) = (k>>1)*ps + m*2 + (k&1)
// so one fragment == one aligned 64-bit load (b64 -> even VGPR pair, no
// reassembly movs). LDS pair strides chosen == 32 (mod 64 banks) so the two
// half-wave K-groups hit disjoint bank halves.
// ---------------------------------------------------------------------------
typedef float v2f __attribute__((ext_vector_type(2)));
typedef float v8f __attribute__((ext_vector_type(8)));

__device__ __forceinline__ v8f wmma4(v2f a, v2f b, v8f c) {
  // 8-arg pattern: (neg_a, A, neg_b, B, c_mod, C, reuse_a, reuse_b)
  return __builtin_amdgcn_wmma_f32_16x16x4_f32(false, a, false, b, (short)0, c,
                                               false, false);
}

// Fragment load from pair-interleaved layout. Caller pre-offsets p to
// (m0/n0, k0): p = base + (k0>>1)*ps + m0*2. Works for A and B fragments.
__device__ __forceinline__ v2f load_frag_pk(const float* __restrict__ p, int ps) {
  int lane = threadIdx.x & 31;
  int m = lane & 15;
  int kp = lane >> 4;
  return *(const v2f*)(p + (size_t)kp * ps + m * 2);
}

// plain C/D store: element (m,n) at p[m*ms + n*ns]
__device__ __forceinline__ void store_frag_c(float* __restrict__ p, int ms, int ns, v8f c) {
  int lane = threadIdx.x & 31;
  int n = lane & 15;
  int mh = (lane >> 4) << 3;
#pragma unroll
  for (int v = 0; v < 8; ++v)
    p[(size_t)(mh + v) * ms + (size_t)n * ns] = c[v];
}

// transposed C store into an n-pair-interleaved layout:
// element (m,n) -> p[(n>>1)*ps + m*2 + (n&1)]  (p pre-offset to (m0, n0))
__device__ __forceinline__ void store_frag_c_tp(float* __restrict__ p, int ps, v8f c) {
  int lane = threadIdx.x & 31;
  int n = lane & 15;
  int mh = (lane >> 4) << 3;
  float* q = p + (size_t)(n >> 1) * ps + (n & 1);
#pragma unroll
  for (int v = 0; v < 8; ++v)
    q[(mh + v) * 2] = c[v];
}

// ---------------------------------------------------------------------------
// Twiddle / reconstruction tables (in ws), K-pair-interleaved:
//  F [128][32] pairs along h (ps=64):  F[h][c] = e^{-2pi i hc/128}
//  G [32][128] pairs along kx (ps=256): G[kx][h] = e^{+2pi i kx h/128}
//  P,Q [32][128] pairs along ky (ps=256): P=s*cos, Q=-s*sin,
//                s = (ky==0 ? 1 : 2)/16384   (Hermitian-W reconstruction)
// ---------------------------------------------------------------------------
__global__ __launch_bounds__(256) void fno_twiddles(float* __restrict__ ws) {
  int idx = blockIdx.x * 256 + threadIdx.x;
  if (idx >= 4096) return;
  float* Fr = ws;
  float* Fi = ws + 4096;
  float* Gr = ws + 8192;
  float* Gi = ws + 12288;
  float* P  = ws + 16384;
  float* Q  = ws + 20480;
  const float w0 = 6.28318530717958647692f / 128.0f;
  {
    int h = idx >> 5, c = idx & 31;
    float th = w0 * (float)((h * c) & 127);
    int a = (h >> 1) * 64 + c * 2 + (h & 1);
    Fr[a] = cosf(th);
    Fi[a] = -sinf(th);
  }
  {
    int r = idx >> 7, w = idx & 127;
    float th = w0 * (float)((r * w) & 127);
    int a = (r >> 1) * 256 + w * 2 + (r & 1);
    Gr[a] = cosf(th);
    Gi[a] = sinf(th);
    float s = (r == 0 ? 1.0f : 2.0f) * (1.0f / 16384.0f);
    P[a] = s * cosf(th);
    Q[a] = -s * sinf(th);
  }
}

// ---------------------------------------------------------------------------
// Forward truncated DFT: one block per (b,c) image.
// T = A*X (M=kx32, K=h128, N=w128); Y = T*F (M=kx32, K=w128, N=ky32)
// Y written to ws layout [h_head][kx][ky][b*16+i] for the mix stage.
// ---------------------------------------------------------------------------
__global__ __launch_bounds__(256) void fno_forward(const float* __restrict__ x,
                                                   const float* __restrict__ Fr,
                                                   const float* __restrict__ Fi,
                                                   float* __restrict__ Yr,
                                                   float* __restrict__ Yi) {
  __shared__ float Xsp[64 * 288];    // image, h-pair interleaved, ps=288
  __shared__ float Ttp[2][64 * 96];  // T^T, w-pair interleaved [w/2][kx][2], ps=96

  const int tid = threadIdx.x;
  const int wave = tid >> 5;
  const int bc = blockIdx.x;
  const int b = bc >> 6, c = bc & 63;
  const int h_head = c >> 4, i_in = c & 15;

  // stage image into LDS, interleaving row pairs: Xsp[(h>>1)*288 + w*2 + (h&1)]
  const float4* xg = (const float4*)(x + (size_t)bc * 16384);
  for (int t = tid; t < 2048; t += 256) {
    int rp = t >> 5, q = t & 31;          // row pair, 4-col group
    float4 a = xg[(2 * rp) * 32 + q];
    float4 bq = xg[(2 * rp + 1) * 32 + q];
    float4 w0 = {a.x, bq.x, a.y, bq.y};
    float4 w1 = {a.z, bq.z, a.w, bq.w};
    *(float4*)&Xsp[rp * 288 + q * 8]     = w0;
    *(float4*)&Xsp[rp * 288 + q * 8 + 4] = w1;
  }
  __syncthreads();

  // ---- stage 1a: T = A * X, A^T = F (pairs along h) ----
  // wave -> (mat, mt) + 4 register-blocked w-tiles: 1 A-frag feeds 4 WMMAs
  {
    const int mat = (wave >> 1) & 1;   // 0: real, 1: imag
    const int mt = wave & 1;           // kx tile
    const int nbase = (wave >> 2) * 4; // w tiles nbase..nbase+3
    const float* At = mat ? Fi : Fr;
    v8f acc[4] = {};
    for (int k0 = 0; k0 < 128; k0 += 4) {
      v2f a = load_frag_pk(At + (k0 >> 1) * 64 + mt * 32, 64);
#pragma unroll
      for (int j = 0; j < 4; ++j) {
        v2f bf = load_frag_pk(Xsp + (k0 >> 1) * 288 + (nbase + j) * 32, 288);
        acc[j] = wmma4(a, bf, acc[j]);
      }
    }
#pragma unroll
    for (int j = 0; j < 4; ++j)  // (m=kx, n=w) -> Ttp[w/2][kx][w&1]
      store_frag_c_tp(&Ttp[mat][(nbase + j) * 8 * 96 + mt * 32], 96, acc[j]);
  }
  __syncthreads();

  // ---- stage 1b: Y = T * F ; 8 tiles (mat x 2mt x 2nt), one per wave ----
  {
    const int mat = wave >> 2;  // 0: Yr = Tr*Fr - Ti*Fi ; 1: Yi = Tr*Fi + Ti*Fr
    const int mt = (wave >> 1) & 1;
    const int nt = wave & 1;
    const float* B1 = mat ? Fi : Fr;  // pairs with Ttp[0] (real T)
    const float* B2 = mat ? Fr : Fi;  // pairs with Ttp[1] (imag T)
    const float sgn = mat ? 1.0f : -1.0f;
    v8f acc = {};
    for (int k0 = 0; k0 < 128; k0 += 4) {
      v2f a1 = load_frag_pk(&Ttp[0][(k0 >> 1) * 96 + mt * 32], 96);
      v2f b1 = load_frag_pk(B1 + (k0 >> 1) * 64 + nt * 32, 64);
      acc = wmma4(a1, b1, acc);
      v2f a2 = load_frag_pk(&Ttp[1][(k0 >> 1) * 96 + mt * 32], 96);
      a2 *= sgn;
      v2f b2 = load_frag_pk(B2 + (k0 >> 1) * 64 + nt * 32, 64);
      acc = wmma4(a2, b2, acc);
    }
    float* dst = (mat ? Yi : Yr) + (size_t)h_head * 524288 +
                 (size_t)(mt * 16) * 16384 + (size_t)(nt * 16) * 512 +
                 (size_t)b * 16 + i_in;
    store_frag_c(dst, 16384, 512, acc);  // (m=kx, n=ky)
  }
}

// ---------------------------------------------------------------------------
// Spectral channel mixing: one wave per (head, kx, ky) mode.
// out[b,o] = sum_i X[b,i]*W[i,o] (complex): M=32(b), N=16(o), K=16(i)
// A-frags are contiguous (i,i+1 adjacent) -> b64; scattered weight frags
// hoisted so each feeds 4 WMMAs. Z written [b][c][kx][ky] for the inverse.
// ---------------------------------------------------------------------------
__global__ __launch_bounds__(256) void fno_mix(const float* __restrict__ Yr,
                                               const float* __restrict__ Yi,
                                               const float* __restrict__ wr,
                                               const float* __restrict__ wi,
                                               float* __restrict__ Zr,
                                               float* __restrict__ Zi) {
  const int wave = threadIdx.x >> 5;
  const int gw = blockIdx.x * 8 + wave;     // 0..4095 = ((h*32+kx)*32+ky)
  const int h = gw >> 10;
  const int kx = (gw >> 5) & 31;
  const int ky = gw & 31;
  const int sp = kx * 32 + ky;
  const int lane = threadIdx.x & 31;
  const int lm = lane & 15;
  const int lkb = (lane >> 4) << 1;

  const float* Xr = Yr + (size_t)gw * 512;  // [b*16+i]
  const float* Xi = Yi + (size_t)gw * 512;
  const float* Wr = wr + (size_t)h * 262144 + sp;  // + (i*16+o)*1024
  const float* Wi = wi + (size_t)h * 262144 + sp;

  v8f accR[2] = {}, accI[2] = {};
#pragma unroll
  for (int k0 = 0; k0 < 16; k0 += 4) {
    v2f bwr, bwi;  // weight fragment (k=i, n=o): gathers, stride 1024/16384
    bwr.x = Wr[(size_t)(k0 + lkb) * 16384 + (size_t)lm * 1024];
    bwr.y = Wr[(size_t)(k0 + lkb + 1) * 16384 + (size_t)lm * 1024];
    bwi.x = Wi[(size_t)(k0 + lkb) * 16384 + (size_t)lm * 1024];
    bwi.y = Wi[(size_t)(k0 + lkb + 1) * 16384 + (size_t)lm * 1024];
    v2f nwi = -bwi;
#pragma unroll
    for (int mt = 0; mt < 2; ++mt) {
      // A-frag (m=b, k=i): i,i+1 contiguous -> single b64 per lane
      v2f axr = *(const v2f*)(Xr + mt * 256 + (size_t)lm * 16 + k0 + lkb);
      v2f axi = *(const v2f*)(Xi + mt * 256 + (size_t)lm * 16 + k0 + lkb);
      accR[mt] = wmma4(axr, bwr, accR[mt]);
      accR[mt] = wmma4(axi, nwi, accR[mt]);
      accI[mt] = wmma4(axr, bwi, accI[mt]);
      accI[mt] = wmma4(axi, bwr, accI[mt]);
    }
  }
#pragma unroll
  for (int mt = 0; mt < 2; ++mt) {
    float* dr = Zr + (size_t)(mt * 16) * 65536 + (size_t)(h * 16) * 1024 + sp;
    float* di = Zi + (size_t)(mt * 16) * 65536 + (size_t)(h * 16) * 1024 + sp;
    store_frag_c(dr, 65536, 1024, accR[mt]);  // (m=b, n=o)
    store_frag_c(di, 65536, 1024, accI[mt]);
  }
}

// ---------------------------------------------------------------------------
// Inverse truncated DFT: one block per (b,c) image.
// C = E*Z (M=h128, K=kx32, N=ky32); out = Cr*P + Ci*Q (M=h128, K=ky32, N=w128)
// ---------------------------------------------------------------------------
__global__ __launch_bounds__(256) void fno_inverse(const float* __restrict__ Zr,
                                                   const float* __restrict__ Zi,
                                                   const float* __restrict__ Gr,
                                                   const float* __restrict__ Gi,
                                                   const float* __restrict__ P,
                                                   const float* __restrict__ Q,
                                                   const float* __restrict__ bias,
                                                   float* __restrict__ out) {
  __shared__ float Zsp[2][16 * 96];    // Z, kx-pair interleaved, ps=96
  __shared__ float Ctp[2][16 * 288];   // C^T, ky-pair interleaved [ky/2][h][2], ps=288

  const int tid = threadIdx.x;
  const int wave = tid >> 5;
  const int bc = blockIdx.x;
  const int c = bc & 63;

  const size_t zoff = (size_t)bc * 1024;
  for (int t = tid; t < 512; t += 256) {
    int kp = t >> 5, ky = t & 31;
    int kx = kp * 2;
    v2f zr, zi;
    zr.x = Zr[zoff + kx * 32 + ky];
    zr.y = Zr[zoff + (kx + 1) * 32 + ky];
    zi.x = Zi[zoff + kx * 32 + ky];
    zi.y = Zi[zoff + (kx + 1) * 32 + ky];
    *(v2f*)&Zsp[0][kp * 96 + ky * 2] = zr;
    *(v2f*)&Zsp[1][kp * 96 + ky * 2] = zi;
  }
  __syncthreads();

  // ---- stage 3a: C = E*Z, E^T = G (pairs along kx) ----
  // wave -> h tile; 4 (Cr/Ci x 2 ky-tiles) tiles register-blocked:
  // 2 A-frag b64 loads (+1 negate) feed 8 WMMAs per k-step.
  {
    const int mt = wave;  // h tile 0..7
    v8f aR[2] = {}, aI[2] = {};
#pragma unroll
    for (int k0 = 0; k0 < 32; k0 += 4) {
      v2f g1 = load_frag_pk(Gr + (k0 >> 1) * 256 + mt * 32, 256);
      v2f g2 = load_frag_pk(Gi + (k0 >> 1) * 256 + mt * 32, 256);
      v2f ng2 = -g2;
#pragma unroll
      for (int nt = 0; nt < 2; ++nt) {
        v2f z0 = load_frag_pk(&Zsp[0][(k0 >> 1) * 96 + nt * 32], 96);
        v2f z1 = load_frag_pk(&Zsp[1][(k0 >> 1) * 96 + nt * 32], 96);
        aR[nt] = wmma4(g1, z0, aR[nt]);   // Cr += Gr'*Zr
        aR[nt] = wmma4(ng2, z1, aR[nt]);  // Cr -= Gi'*Zi
        aI[nt] = wmma4(g1, z1, aI[nt]);   // Ci += Gr'*Zi
        aI[nt] = wmma4(g2, z0, aI[nt]);   // Ci += Gi'*Zr
      }
    }
#pragma unroll
    for (int nt = 0; nt < 2; ++nt) {  // (m=h, n=ky) -> Ctp[ky/2][h][ky&1]
      store_frag_c_tp(&Ctp[0][nt * 8 * 288 + mt * 32], 288, aR[nt]);
      store_frag_c_tp(&Ctp[1][nt * 8 * 288 + mt * 32], 288, aI[nt]);
    }
  }
  __syncthreads();

  // ---- stage 3b: out = Cr*P + Ci*Q ----
  // wave -> h tile; 8 w-tiles register-blocked: 2 LDS b64 A-frag loads feed
  // 16 WMMAs per k-step.
  {
    const int mt = wave;
    const float bv = bias[c];
    const int lane = tid & 31;
    const int ncol = lane & 15;
    const int mh = (lane >> 4) << 3;
    v8f acc[8] = {};
#pragma unroll
    for (int k0 = 0; k0 < 32; k0 += 4) {
      v2f a1 = load_frag_pk(&Ctp[0][(k0 >> 1) * 288 + mt * 32], 288);
      v2f a2 = load_frag_pk(&Ctp[1][(k0 >> 1) * 288 + mt * 32], 288);
#pragma unroll
      for (int nt = 0; nt < 8; ++nt) {
        v2f b1 = load_frag_pk(P + (k0 >> 1) * 256 + nt * 32, 256);
        v2f b2 = load_frag_pk(Q + (k0 >> 1) * 256 + nt * 32, 256);
        acc[nt] = wmma4(a1, b1, acc[nt]);
        acc[nt] = wmma4(a2, b2, acc[nt]);
      }
    }
#pragma unroll
    for (int nt = 0; nt < 8; ++nt) {
      float* op = out + (size_t)bc * 16384 + (size_t)(mt * 16) * 128 + nt * 16;
#pragma unroll
      for (int v = 0; v < 8; ++v)
        op[(mh + v) * 128 + ncol] = acc[nt][v] + bv;
    }
  }
}

// ---------------------------------------------------------------------------
// ws layout (floats): [Fr 4096][Fi 4096][Gr 4096][Gi 4096][P 4096][Q 4096]
//                     [Yr 2097152][Yi 2097152][Zr 2097152][Zi 2097152]
// total ~33.7 MB
// ---------------------------------------------------------------------------
extern "C" void kernel_launch(void* const* d_in, const int* in_sizes, int n_in,
                              void* d_out, int out_size, void* d_ws, size_t ws_size,
                              hipStream_t stream) {
  const float* x    = (const float*)d_in[0];
  const float* wr   = (const float*)d_in[1];
  const float* wi   = (const float*)d_in[2];
  const float* bias = (const float*)d_in[3];
  float* out = (float*)d_out;

  float* ws = (float*)d_ws;
  float* Fr = ws;
  float* Fi = ws + 4096;
  float* Gr = ws + 8192;
  float* Gi = ws + 12288;
  float* P  = ws + 16384;
  float* Q  = ws + 20480;
  float* Yr = ws + 24576;
  float* Yi = Yr + 2097152;
  float* Zr = Yi + 2097152;
  float* Zi = Zr + 2097152;

  fno_twiddles<<<dim3(16), dim3(256), 0, stream>>>(ws);
  fno_forward<<<dim3(2048), dim3(256), 0, stream>>>(x, Fr, Fi, Yr, Yi);
  fno_mix<<<dim3(512), dim3(256), 0, stream>>>(Yr, Yi, wr, wi, Zr, Zi);
  fno_inverse<<<dim3(2048), dim3(256), 0, stream>>>(Zr, Zi, Gr, Gi, P, Q, bias, out);
}